// Dual_Attention_32246614458842
// MI455X (gfx1250) — compile-verified
//
#include <hip/hip_runtime.h>
#include <cstdint>
#include <cstddef>

// ---------------- problem constants (from reference) ----------------
#define NCN   15000
#define NVN   15000
#define EDG   150000
#define DD    128     // model dim
#define HH    8       // heads
#define DKV   16      // per-head dim (DK == DV == 16)
#define DFF   512     // FFN hidden
#define EPSF  1e-8f
#define LNEPS 1e-5f

typedef __attribute__((ext_vector_type(2))) float v2f;
typedef __attribute__((ext_vector_type(8))) float v8f;

// =====================================================================
// Generic fp32 WMMA GEMM:  C[N x Dout] = act( A[N x Din] * W[Din x Dout] + bias )
// A may be a virtual concat of A (DinA cols) and A2 (Din-DinA cols).
// One wave computes a 16x16 tile via V_WMMA_F32_16X16X4_F32 (K-step 4).
// blockDim = 128 (4 waves): block tile = 16 rows x 64 cols.
// A tile is staged into LDS with CDNA5 async global->LDS B128 copies
// (ASYNCcnt), bypassing VGPRs; completion via s_wait_asynccnt + barrier.
// Fragment layouts per CDNA5 ISA 7.12.2:
//   A 16x4 : lane l -> m = l&15, holds K = 2*(l>>4) and 2*(l>>4)+1
//   B 4x16 : lane l -> n = l&15, holds K rows 2*(l>>4), 2*(l>>4)+1
//   C 16x16: vgpr v, lane l -> row = v + 8*(l>>4), col = l&15
// =====================================================================
__global__ void gemm_wmma_kernel(const float* __restrict__ A,
                                 const float* __restrict__ A2,
                                 const float* __restrict__ W,
                                 const float* __restrict__ bias,
                                 float* __restrict__ C,
                                 int N, int Din, int DinA, int Dout, int act)
{
    extern __shared__ float As[];                 // 16 x Din tile of A (LDS offset 0)
    const int tid  = threadIdx.x;
    const int lane = tid & 31;
    const int wave = tid >> 5;
    const int row0 = blockIdx.y * 16;
    const int col0 = blockIdx.x * 64 + wave * 16;

    // ---- async stage of the A tile into LDS, 16 bytes per issue ----
    // All A operands have row strides that are multiples of 4 floats and
    // 16B-aligned bases, and Din % 4 == 0, so B128 chunks are legal and
    // never straddle the A/A2 concat boundary (DinA % 4 == 0).
    const int chunks = 4 * Din;                   // (16*Din)/4 chunks of 4 floats
    for (int j = tid; j < chunks; j += blockDim.x) {
        const int i = j * 4;
        const int r = i / Din, k = i - r * Din;
        const int row = row0 + r;
        if (row < N) {
            const float* gp = (k < DinA)
                ? &A[(size_t)row * DinA + k]
                : &A2[(size_t)row * (Din - DinA) + (k - DinA)];
            const unsigned ldsoff = (unsigned)(i * 4);   // byte offset of As[i]
            asm volatile("global_load_async_to_lds_b128 %0, %1, off"
                         :: "v"(ldsoff), "v"(gp) : "memory");
        } else {
            As[i] = 0.f; As[i + 1] = 0.f; As[i + 2] = 0.f; As[i + 3] = 0.f;
        }
    }
    asm volatile("s_wait_asynccnt 0x0" ::: "memory");   // own wave's async copies done
    __syncthreads();                                    // all waves' copies visible

    v8f acc = {};
    const int m  = lane & 15;
    const int kb = (lane >> 4) * 2;
    const int n  = lane & 15;
    const float* As_m = As + m * Din;

    for (int k = 0; k < Din; k += 4) {
        v2f a, b;
        a.x = As_m[k + kb];
        a.y = As_m[k + kb + 1];
        const float* Wp = W + (size_t)(k + kb) * Dout + col0 + n;
        b.x = Wp[0];
        b.y = Wp[Dout];
        acc = __builtin_amdgcn_wmma_f32_16x16x4_f32(
            /*neg_a=*/false, a, /*neg_b=*/false, b,
            /*c_mod=*/(short)0, acc, /*reuse_a=*/false, /*reuse_b=*/false);
    }

    const int col = col0 + n;
    const float bv = bias ? bias[col] : 0.f;
#pragma unroll
    for (int v = 0; v < 8; ++v) {
        int row = row0 + v + (lane >> 4) * 8;
        if (row < N) {
            float x = acc[v] + bv;
            if (act == 1)      x = fmaxf(x, 0.f);            // relu
            else if (act == 2) x = 1.f / (1.f + __expf(-x)); // sigmoid
            C[(size_t)row * Dout + col] = x;
        }
    }
}

// ---------------- small-K input embedding: Y = relu(X*W1 + b1) ----------------
__global__ void linear_small_relu_kernel(const float* __restrict__ X,
                                         const float* __restrict__ W,
                                         const float* __restrict__ b,
                                         float* __restrict__ Y, int N, int Kin)
{
    int idx = blockIdx.x * blockDim.x + threadIdx.x;
    if (idx >= N * DD) return;
    int nrow = idx >> 7;
    int col  = idx & 127;
    const float* x = X + (size_t)nrow * Kin;
    float s = b[col];
    for (int k = 0; k < Kin; ++k) s += x[k] * W[k * DD + col];
    Y[idx] = fmaxf(s, 0.f);
}

// ---------------- linear attention: KV = sum_n K[n] (x) V[n], Ksum = sum_n K[n] ----
__global__ void kv_reduce_kernel(const float* __restrict__ Km,
                                 const float* __restrict__ Vm,
                                 float* __restrict__ KV,     // [HH*16*16]
                                 float* __restrict__ Ksum,   // [HH*16]
                                 int N)
{
    __shared__ float Kr[DD], Vr[DD];
    const int t = threadIdx.x;        // 128: owns (head h, k-dim kd)
    const int h = t >> 4, kd = t & 15;
    const int chunk = (N + gridDim.x - 1) / gridDim.x;
    const int n0 = blockIdx.x * chunk;
    const int n1 = (n0 + chunk < N) ? n0 + chunk : N;

    float kv[16];
#pragma unroll
    for (int v = 0; v < 16; ++v) kv[v] = 0.f;
    float ks = 0.f;

    for (int nrow = n0; nrow < n1; ++nrow) {
        Kr[t] = Km[(size_t)nrow * DD + t];
        Vr[t] = Vm[(size_t)nrow * DD + t];
        __syncthreads();
        float kk = Kr[h * 16 + kd];
        ks += kk;
#pragma unroll
        for (int v = 0; v < 16; ++v) kv[v] += kk * Vr[h * 16 + v];
        __syncthreads();
    }
    float* kvp = KV + (size_t)(h * 16 + kd) * 16;
#pragma unroll
    for (int v = 0; v < 16; ++v) atomicAdd(&kvp[v], kv[v]);
    atomicAdd(&Ksum[h * 16 + kd], ks);
}

// ---------------- O[n] = (Q[n] @ KV) / (Q[n] . Ksum + eps), per head ----------------
__global__ void lin_attn_apply_kernel(const float* __restrict__ Qm,
                                      const float* __restrict__ KV,
                                      const float* __restrict__ Ksum,
                                      float* __restrict__ O, int N)
{
    __shared__ float kvs[HH * DKV * DKV];
    __shared__ float kss[DD];
    __shared__ float qr[DD];
    const int t = threadIdx.x;  // 128: owns (h, v)
    for (int i = t; i < HH * DKV * DKV; i += 128) kvs[i] = KV[i];
    kss[t] = Ksum[t];
    __syncthreads();
    const int h = t >> 4, v = t & 15;
    const int chunk = (N + gridDim.x - 1) / gridDim.x;
    const int n0 = blockIdx.x * chunk;
    const int n1 = (n0 + chunk < N) ? n0 + chunk : N;

    for (int nrow = n0; nrow < n1; ++nrow) {
        qr[t] = Qm[(size_t)nrow * DD + t];
        __syncthreads();
        float num = 0.f, den = 0.f;
#pragma unroll
        for (int k = 0; k < 16; ++k) {
            float q = qr[h * 16 + k];
            num += q * kvs[(h * 16 + k) * 16 + v];
            den += q * kss[h * 16 + k];
        }
        O[(size_t)nrow * DD + t] = num / (den + EPSF);
        __syncthreads();
    }
}

// ---------------- Ee coefficients: A = ee_w @ we, B = ee_b @ we (each 128) ---------
__global__ void ee_coef_kernel(const float* __restrict__ ee_w,
                               const float* __restrict__ ee_b,
                               const float* __restrict__ we,
                               float* __restrict__ Ac, float* __restrict__ Bc)
{
    int d = threadIdx.x;   // 128
    float a = 0.f, b = 0.f;
    for (int k = 0; k < DD; ++k) {
        float w = we[k * DD + d];
        a += ee_w[k] * w;
        b += ee_b[k] * w;
    }
    Ac[d] = a;
    Bc[d] = b;
}

// ---------------- edge scatter: thread = (edge, head) ----------------
__global__ void cross_edge_kernel(const int* __restrict__ qIdx,
                                  const int* __restrict__ kvIdx,
                                  const float* __restrict__ ef,   // scalar per edge
                                  const float* __restrict__ Qm,
                                  const float* __restrict__ Km,
                                  const float* __restrict__ Vm,
                                  const float* __restrict__ Ac,
                                  const float* __restrict__ Bc,
                                  float* __restrict__ Oacc,       // [Nq x 128]
                                  float* __restrict__ coeff)      // [Nq x 8]
{
    __shared__ float As_s[DD], Bs_s[DD];
    if (threadIdx.x < DD) {
        As_s[threadIdx.x] = Ac[threadIdx.x];
        Bs_s[threadIdx.x] = Bc[threadIdx.x];
    }
    __syncthreads();

    int idx = blockIdx.x * blockDim.x + threadIdx.x;
    if (idx >= EDG * HH) return;
    const int e = idx >> 3, h = idx & 7;
    const int qn = qIdx[e], kn = kvIdx[e];
    const float es = ef[e];
    const float* q  = Qm + (size_t)qn * DD + h * 16;
    const float* k  = Km + (size_t)kn * DD + h * 16;
    const float* vv = Vm + (size_t)kn * DD + h * 16;

    float s = 0.f;
#pragma unroll
    for (int d = 0; d < 16; ++d) {
        float ee = es * As_s[h * 16 + d] + Bs_s[h * 16 + d];
        s += q[d] * k[d] * ee;
    }
    s *= 0.25f;                               // 1/sqrt(DK=16)
    s = fminf(fmaxf(s, -5.f), 5.f);
    const float attn = __expf(s);

    float* oa = Oacc + (size_t)qn * DD + h * 16;
#pragma unroll
    for (int d = 0; d < 16; ++d) atomicAdd(&oa[d], attn * vv[d]);
    atomicAdd(&coeff[(size_t)qn * HH + h], attn);
}

// ---------------- normalize scattered messages ----------------
__global__ void cross_norm_kernel(const float* __restrict__ Oacc,
                                  const float* __restrict__ coeff,
                                  float* __restrict__ O, int N)
{
    int idx = blockIdx.x * blockDim.x + threadIdx.x;
    if (idx >= N * DD) return;
    int nrow = idx >> 7, d = idx & 127, h = d >> 4;
    O[idx] = Oacc[idx] / (coeff[(size_t)nrow * HH + h] + EPSF);
}

// ---------------- Y = LN(X (+ R)) * g + b ; one wave32 per row, 4 vals/lane -------
__global__ void residual_ln_kernel(const float* __restrict__ X,
                                   const float* __restrict__ R,
                                   const float* __restrict__ g,
                                   const float* __restrict__ b,
                                   float* __restrict__ Y, int N, int has_res)
{
    const int lane = threadIdx.x & 31;
    const int wave = threadIdx.x >> 5;
    const int row = blockIdx.x * 4 + wave;
    if (row >= N) return;
    const float* x = X + (size_t)row * DD;
    const float* r = R + (size_t)row * DD;

    float v[4];
#pragma unroll
    for (int i = 0; i < 4; ++i) {
        int c = lane + i * 32;
        v[i] = x[c] + (has_res ? r[c] : 0.f);
    }
    float s  = v[0] + v[1] + v[2] + v[3];
    float s2 = v[0]*v[0] + v[1]*v[1] + v[2]*v[2] + v[3]*v[3];
#pragma unroll
    for (int off = 16; off > 0; off >>= 1) {
        s  += __shfl_xor(s,  off, 32);
        s2 += __shfl_xor(s2, off, 32);
    }
    const float mean = s * (1.f / 128.f);
    const float var  = s2 * (1.f / 128.f) - mean * mean;
    const float rstd = rsqrtf(var + LNEPS);
#pragma unroll
    for (int i = 0; i < 4; ++i) {
        int c = lane + i * 32;
        Y[(size_t)row * DD + c] = (v[i] - mean) * rstd * g[c] + b[c];
    }
}

// ---------------- utility ----------------
__global__ void fill_kernel(float* __restrict__ p, size_t n, float v)
{
    size_t i = (size_t)blockIdx.x * blockDim.x + threadIdx.x;
    if (i < n) p[i] = v;
}
__global__ void copy_kernel(const float* __restrict__ s, float* __restrict__ d, size_t n)
{
    size_t i = (size_t)blockIdx.x * blockDim.x + threadIdx.x;
    if (i < n) d[i] = s[i];
}

// =====================================================================
// host launcher
// =====================================================================
extern "C" void kernel_launch(void* const* d_in, const int* in_sizes, int n_in,
                              void* d_out, int out_size, void* d_ws, size_t ws_size,
                              hipStream_t stream)
{
    // ---- inputs (setup_inputs dict order) ----
    const float* cf    = (const float*)d_in[0];      // (NC,5)
    const int*   eidx  = (const int*)  d_in[1];      // (2,E)
    const float* efeat = (const float*)d_in[2];      // (E,)
    const float* vf    = (const float*)d_in[3];      // (NV,19)

    // ---- params: jax tree leaves, dict keys alphabetical, depth-first ----
    int p = 4;
    // emb: ce_b1, ce_b2, ce_w1, ce_w2, ee_b, ee_w, ve_b1, ve_b2, ve_w1, ve_w2
    const float* ce_b1 = (const float*)d_in[p++];
    const float* ce_b2 = (const float*)d_in[p++];
    const float* ce_w1 = (const float*)d_in[p++];
    const float* ce_w2 = (const float*)d_in[p++];
    const float* ee_b  = (const float*)d_in[p++];
    const float* ee_w  = (const float*)d_in[p++];
    const float* ve_b1 = (const float*)d_in[p++];
    const float* ve_b2 = (const float*)d_in[p++];
    const float* ve_w1 = (const float*)d_in[p++];
    const float* ve_w2 = (const float*)d_in[p++];
    // layer keys sorted: ca_c2v, ca_v2c, ffn_c, ffn_v, fus_c, fus_v, sa_con, sa_var
    // cross attn keys sorted: fc, ln_b, ln_g, we, wk, wq, wv
    const float* c2v_fc  = (const float*)d_in[p++];
    const float* c2v_lnb = (const float*)d_in[p++];
    const float* c2v_lng = (const float*)d_in[p++];
    const float* c2v_we  = (const float*)d_in[p++];
    const float* c2v_wk  = (const float*)d_in[p++];
    const float* c2v_wq  = (const float*)d_in[p++];
    const float* c2v_wv  = (const float*)d_in[p++];
    const float* v2c_fc  = (const float*)d_in[p++];
    const float* v2c_lnb = (const float*)d_in[p++];
    const float* v2c_lng = (const float*)d_in[p++];
    const float* v2c_we  = (const float*)d_in[p++];
    const float* v2c_wk  = (const float*)d_in[p++];
    const float* v2c_wq  = (const float*)d_in[p++];
    const float* v2c_wv  = (const float*)d_in[p++];
    // ffn keys sorted: ln_b, ln_g, w1, w2
    const float* ffnc_lnb = (const float*)d_in[p++];
    const float* ffnc_lng = (const float*)d_in[p++];
    const float* ffnc_w1  = (const float*)d_in[p++];
    const float* ffnc_w2  = (const float*)d_in[p++];
    const float* ffnv_lnb = (const float*)d_in[p++];
    const float* ffnv_lng = (const float*)d_in[p++];
    const float* ffnv_w1  = (const float*)d_in[p++];
    const float* ffnv_w2  = (const float*)d_in[p++];
    // fusion keys sorted: b, ln_b, ln_g, w
    const float* fusc_b   = (const float*)d_in[p++];
    const float* fusc_lnb = (const float*)d_in[p++];
    const float* fusc_lng = (const float*)d_in[p++];
    const float* fusc_w   = (const float*)d_in[p++];
    const float* fusv_b   = (const float*)d_in[p++];
    const float* fusv_lnb = (const float*)d_in[p++];
    const float* fusv_lng = (const float*)d_in[p++];
    const float* fusv_w   = (const float*)d_in[p++];
    // self attn keys sorted: fc, ln_b, ln_g, wk, wq, wv
    const float* sac_fc  = (const float*)d_in[p++];
    const float* sac_lnb = (const float*)d_in[p++];
    const float* sac_lng = (const float*)d_in[p++];
    const float* sac_wk  = (const float*)d_in[p++];
    const float* sac_wq  = (const float*)d_in[p++];
    const float* sac_wv  = (const float*)d_in[p++];
    const float* sav_fc  = (const float*)d_in[p++];
    const float* sav_lnb = (const float*)d_in[p++];
    const float* sav_lng = (const float*)d_in[p++];
    const float* sav_wk  = (const float*)d_in[p++];
    const float* sav_wq  = (const float*)d_in[p++];
    const float* sav_wv  = (const float*)d_in[p++];

    const int* srcI = eidx;        // con node per edge
    const int* tgtI = eidx + EDG;  // var node per edge

    // ---- workspace carve-up (floats) ----
    const size_t NF = (size_t)NVN * DD;
    float* ws = (float*)d_ws;
    float* con = ws;             ws += NF;
    float* var = ws;             ws += NF;
    float* sc  = ws;             ws += NF;   // self-attn con out
    float* cc  = ws;             ws += NF;   // cross-attn con out
    float* sv  = ws;             ws += NF;
    float* cv  = ws;             ws += NF;
    float* Qb  = ws;             ws += NF;
    float* Kb  = ws;             ws += NF;
    float* Vb  = ws;             ws += NF;
    float* T1  = ws;             ws += NF;   // also edge Oacc
    float* T2  = ws;             ws += NF;
    float* Hb  = ws;             ws += (size_t)NVN * DFF;
    float* KVb = ws;             ws += HH * DKV * DKV;   // contiguous with KSb
    float* KSb = ws;             ws += DD;
    float* Ac  = ws;             ws += DD;
    float* Bc  = ws;             ws += DD;
    float* cof = ws;             ws += (size_t)NVN * HH;
    (void)ws_size; (void)n_in; (void)in_sizes; (void)out_size;

    auto gemm = [&](const float* A, const float* A2, int DinA, int Din,
                    const float* W, const float* bias, float* C,
                    int N, int Dout, int act) {
        dim3 grid(Dout / 64, (N + 15) / 16);
        size_t lds = (size_t)16 * Din * sizeof(float);
        gemm_wmma_kernel<<<grid, 128, lds, stream>>>(A, A2, W, bias, C,
                                                     N, Din, DinA, Dout, act);
    };
    auto fill0 = [&](float* ptr, size_t n) {
        fill_kernel<<<(int)((n + 255) / 256), 256, 0, stream>>>(ptr, n, 0.f);
    };
    auto ln = [&](const float* X, const float* R, const float* g, const float* b,
                  float* Y, int N, int has_res) {
        residual_ln_kernel<<<(N + 3) / 4, 128, 0, stream>>>(X, R, g, b, Y, N, has_res);
    };

    auto self_attn = [&](const float* x, int N,
                         const float* wq, const float* wk, const float* wv,
                         const float* fc, const float* lng, const float* lnb,
                         float* out) {
        gemm(x, nullptr, DD, DD, wq, nullptr, Qb, N, DD, 2);  // sigmoid
        gemm(x, nullptr, DD, DD, wk, nullptr, Kb, N, DD, 2);  // sigmoid
        gemm(x, nullptr, DD, DD, wv, nullptr, Vb, N, DD, 0);
        fill0(KVb, HH * DKV * DKV + DD);                      // KV + Ksum
        kv_reduce_kernel<<<60, 128, 0, stream>>>(Kb, Vb, KVb, KSb, N);
        lin_attn_apply_kernel<<<120, 128, 0, stream>>>(Qb, KVb, KSb, T2, N);
        gemm(T2, nullptr, DD, DD, fc, nullptr, Qb, N, DD, 0); // out @ fc
        ln(Qb, x, lng, lnb, out, N, 1);                       // LN(.. + x)
    };

    auto cross_attn = [&](const float* xq, int Nq, const float* xkv,
                          const int* qidx, const int* kvidx,
                          const float* wq, const float* wk, const float* wv,
                          const float* we, const float* fc,
                          const float* lng, const float* lnb, float* out,
                          int Nkv) {
        gemm(xq,  nullptr, DD, DD, wq, nullptr, Qb, Nq,  DD, 0);
        gemm(xkv, nullptr, DD, DD, wk, nullptr, Kb, Nkv, DD, 0);
        gemm(xkv, nullptr, DD, DD, wv, nullptr, Vb, Nkv, DD, 0);
        ee_coef_kernel<<<1, 128, 0, stream>>>(ee_w, ee_b, we, Ac, Bc);
        fill0(T1, (size_t)Nq * DD);
        fill0(cof, (size_t)Nq * HH);
        cross_edge_kernel<<<(EDG * HH + 255) / 256, 256, 0, stream>>>(
            qidx, kvidx, efeat, Qb, Kb, Vb, Ac, Bc, T1, cof);
        cross_norm_kernel<<<(Nq * DD + 255) / 256, 256, 0, stream>>>(T1, cof, T2, Nq);
        gemm(T2, nullptr, DD, DD, fc, nullptr, Qb, Nq, DD, 0);
        ln(Qb, xq, lng, lnb, out, Nq, 1);
    };

    auto fus_ffn = [&](const float* a, const float* b2, int N,
                       const float* fw, const float* fb,
                       const float* flng, const float* flnb,
                       const float* w1, const float* w2,
                       const float* lng2, const float* lnb2, float* out) {
        gemm(a, b2, DD, 2 * DD, fw, fb, T1, N, DD, 1);         // relu(concat@w + b)
        ln(T1, T1, flng, flnb, T2, N, 0);                      // fusion LN -> h
        gemm(T2, nullptr, DD, DD, w1, nullptr, Hb, N, DFF, 1); // relu(h@w1)
        gemm(Hb, nullptr, DFF, DFF, w2, nullptr, T1, N, DD, 0);
        ln(T1, T2, lng2, lnb2, out, N, 1);                     // LN(.. + h)
    };

    // ---- embeddings ----
    linear_small_relu_kernel<<<(NCN * DD + 127) / 128, 128, 0, stream>>>(
        cf, ce_w1, ce_b1, T1, NCN, 5);
    gemm(T1, nullptr, DD, DD, ce_w2, ce_b2, con, NCN, DD, 0);
    linear_small_relu_kernel<<<(NVN * DD + 127) / 128, 128, 0, stream>>>(
        vf, ve_w1, ve_b1, T1, NVN, 19);
    gemm(T1, nullptr, DD, DD, ve_w2, ve_b2, var, NVN, DD, 0);

    // ---- layer (all four branches read the pre-update con/var) ----
    self_attn(var, NVN, sav_wq, sav_wk, sav_wv, sav_fc, sav_lng, sav_lnb, sv);
    cross_attn(var, NVN, con, tgtI, srcI,
               v2c_wq, v2c_wk, v2c_wv, v2c_we, v2c_fc, v2c_lng, v2c_lnb, cv, NCN);
    self_attn(con, NCN, sac_wq, sac_wk, sac_wv, sac_fc, sac_lng, sac_lnb, sc);
    cross_attn(con, NCN, var, srcI, tgtI,
               c2v_wq, c2v_wk, c2v_wv, c2v_we, c2v_fc, c2v_lng, c2v_lnb, cc, NVN);

    fus_ffn(sc, cc, NCN, fusc_w, fusc_b, fusc_lng, fusc_lnb,
            ffnc_w1, ffnc_w2, ffnc_lng, ffnc_lnb, con);
    fus_ffn(sv, cv, NVN, fusv_w, fusv_b, fusv_lng, fusv_lnb,
            ffnv_w1, ffnv_w2, ffnv_lng, ffnv_lnb, var);

    // ---- output: (var, con) concatenated ----
    float* out = (float*)d_out;
    copy_kernel<<<(int)((NF + 255) / 256), 256, 0, stream>>>(var, out, NF);
    copy_kernel<<<(int)((NF + 255) / 256), 256, 0, stream>>>(con, out + NF, NF);
}